// EntityGraphRefinement_29497835389105
// MI455X (gfx1250) — compile-verified
//
#include <hip/hip_runtime.h>
#include <hip/hip_bf16.h>

// ---------------------------------------------------------------------------
// EntityGraphRefinement — MI455X (gfx1250) implementation.
//
// Key insight: softmax rows sum to 1 over S, so activity = 1/S = 0.00195 < 0.1
// => entity_mask == 0 and graph == 0 always. Only `entities` needs real work:
//   x = emb + pos
//   q = EL @ Wq ; k = x @ Wk ; v = x @ Wv            (fp32 WMMA GEMMs)
//   scores = q @ k^T / sqrt(32) ; softmax ; ctx = attn @ v   (batched WMMA)
//   entities = ctx @ Wo                              (fp32 WMMA GEMM)
// ---------------------------------------------------------------------------

typedef __attribute__((ext_vector_type(2))) float v2f;
typedef __attribute__((ext_vector_type(8))) float v8f;

#define BCONST 8
#define SCONST 512
#define DCONST 256
#define MCONST 64
#define HCONST 8
#define DHCONST 32

// ---------------------------------------------------------------------------
// x[b,s,d] = emb[b,s,d] + pos_enc[s,d]
// ---------------------------------------------------------------------------
__global__ void add_pos_kernel(const float* __restrict__ emb,
                               const float* __restrict__ pos,
                               float* __restrict__ x, int total) {
    int i = blockIdx.x * blockDim.x + threadIdx.x;
    if (i < total) {
        int sd = i % (SCONST * DCONST);   // pos rows 0..S-1 are contiguous
        x[i] = emb[i] + pos[sd];
    }
}

__global__ void zero_f32_kernel(float* __restrict__ p, int n) {
    int i = blockIdx.x * blockDim.x + threadIdx.x;
    if (i < n) p[i] = 0.0f;
}

// ---------------------------------------------------------------------------
// Fragment layouts for V_WMMA_F32_16X16X4_F32 (wave32):
//   A frag:  lane = khalf*16 + m,   vgpr r -> A[m][k0 + khalf*2 + r]
//   B frag:  lane = khalf*16 + n,   vgpr r -> B[k0 + khalf*2 + r][n]
//   C frag:  vgpr v, lane l        -> C[v + 8*(l>=16)][l%16]
// Batch offset for z = blockIdx.z:  off = (z/zdiv)*s0 + (z%zdiv)*s1
// ---------------------------------------------------------------------------

// ---------------------------------------------------------------------------
// NN GEMM (row-major B, compile-time LDB): C = alpha * A @ B.
// Block = 128 threads = 4 wave32, computing a 16x128 C tile. The block's B
// panel is staged chunk-by-chunk (32 K-rows x 128 cols = 16 KB) into LDS with
// GLOBAL_LOAD_ASYNC_TO_LDS_B128 (ASYNCcnt), then fragments come from DS loads.
// B panels are reused across every M-tile, so staging cuts global traffic 4x.
// ---------------------------------------------------------------------------
template <int LDB>
__global__ __launch_bounds__(128)
void wmma_gemm_nn(const float* __restrict__ A, int lda, int zdivA, long long sA0, long long sA1,
                  const float* __restrict__ B, int zdivB, long long sB0, long long sB1,
                  float* __restrict__ C,       int ldc, int zdivC, long long sC0, long long sC1,
                  int M, int N, int K, float alpha) {
    __shared__ float ldsB[32 * 128];

    const int z    = blockIdx.z;
    const int tid  = threadIdx.x;
    const int lane = tid & 31;
    const int wave = tid >> 5;

    const int row0    = blockIdx.y * 16;
    const int colBase = blockIdx.x * 128;
    const int col0    = colBase + wave * 32;          // 16x32 sub-tile per wave
    const bool active = (row0 < M) && (col0 < N);     // wave-uniform

    A += (long long)(z / zdivA) * sA0 + (long long)(z % zdivA) * sA1;
    B += (long long)(z / zdivB) * sB0 + (long long)(z % zdivB) * sB1;
    C += (long long)(z / zdivC) * sC0 + (long long)(z % zdivC) * sC1;

    const int mlane = lane & 15;
    const int khalf = lane >> 4;

    const float* ap = A + (long long)((active ? row0 : 0) + mlane) * lda + khalf * 2;
    __builtin_prefetch(ap, 0, 0);                     // global_prefetch_b8

    const unsigned ldsBase = (unsigned)(unsigned long long)(&ldsB[0]); // flat->LDS: low 32 bits

    v8f acc0 = {};
    v8f acc1 = {};

    for (int kc = 0; kc < K; kc += 32) {
        // --- cooperative async stage of B[kc..kc+31][colBase..colBase+127] ---
        #pragma unroll
        for (int i = 0; i < 8; ++i) {
            int f4  = i * 128 + tid;                  // float4 index in chunk
            int row = f4 >> 5;
            int c4  = f4 & 31;
            const float* g = B + (long long)(kc + row) * LDB + colBase + c4 * 4;
            unsigned laddr = ldsBase + (unsigned)f4 * 16u;
            asm volatile("global_load_async_to_lds_b128 %0, %1, off"
                         :: "v"(laddr), "v"(g) : "memory");
        }
        asm volatile("s_wait_asynccnt 0x0" ::: "memory");
        __syncthreads();

        if (active) {
            #pragma unroll
            for (int kk = 0; kk < 32; kk += 4) {
                v2f a = *(const v2f*)(ap + kc + kk);
                int rb = (kk + khalf * 2) * 128 + wave * 32 + mlane;
                v2f b0, b1;
                b0.x = ldsB[rb];
                b0.y = ldsB[rb + 128];
                b1.x = ldsB[rb + 16];
                b1.y = ldsB[rb + 144];
                acc0 = __builtin_amdgcn_wmma_f32_16x16x4_f32(false, a, false, b0,
                                                             (short)0, acc0, false, false);
                acc1 = __builtin_amdgcn_wmma_f32_16x16x4_f32(false, a, false, b1,
                                                             (short)0, acc1, false, false);
            }
        }
        __syncthreads();                              // chunk consumed; safe to overwrite
    }

    if (active) {
        #pragma unroll
        for (int v = 0; v < 8; ++v) {
            int mr = row0 + khalf * 8 + v;
            C[(long long)mr * ldc + col0 + mlane]      = alpha * acc0[v];
            C[(long long)mr * ldc + col0 + 16 + mlane] = alpha * acc1[v];
        }
    }
}

// ---------------------------------------------------------------------------
// NT GEMM (B supplied transposed: Bt[n][k], row-major, compile-time LDB):
// C = alpha * A @ Bt^T. Fragments of both A and Bt are contiguous b64 loads,
// so this streams directly from global/L2. 16x32 C tile per wave.
// ---------------------------------------------------------------------------
template <int LDB>
__global__ __launch_bounds__(128)
void wmma_gemm_nt(const float* __restrict__ A, int lda, int zdivA, long long sA0, long long sA1,
                  const float* __restrict__ B, int zdivB, long long sB0, long long sB1,
                  float* __restrict__ C,       int ldc, int zdivC, long long sC0, long long sC1,
                  int M, int N, int K, float alpha) {
    const int z    = blockIdx.z;
    const int lane = threadIdx.x & 31;
    const int wave = threadIdx.x >> 5;

    const int row0 = blockIdx.y * 16;
    const int col0 = (blockIdx.x * 4 + wave) * 32;
    if (row0 >= M || col0 >= N) return;               // wave-uniform branch

    A += (long long)(z / zdivA) * sA0 + (long long)(z % zdivA) * sA1;
    B += (long long)(z / zdivB) * sB0 + (long long)(z % zdivB) * sB1;
    C += (long long)(z / zdivC) * sC0 + (long long)(z % zdivC) * sC1;

    const int mlane = lane & 15;
    const int khalf = lane >> 4;

    const float* ap  = A + (long long)(row0 + mlane) * lda + khalf * 2;
    const float* bp0 = B + (long long)(col0 + mlane) * LDB + khalf * 2;
    const float* bp1 = bp0 + 16 * LDB;
    __builtin_prefetch(ap, 0, 0);
    __builtin_prefetch(bp0, 0, 0);

    v8f acc0 = {};
    v8f acc1 = {};
    #pragma unroll 8
    for (int k = 0; k < K; k += 4) {
        v2f a  = *(const v2f*)(ap + k);
        v2f b0 = *(const v2f*)(bp0 + k);
        v2f b1 = *(const v2f*)(bp1 + k);
        acc0 = __builtin_amdgcn_wmma_f32_16x16x4_f32(false, a, false, b0,
                                                     (short)0, acc0, false, false);
        acc1 = __builtin_amdgcn_wmma_f32_16x16x4_f32(false, a, false, b1,
                                                     (short)0, acc1, false, false);
    }

    #pragma unroll
    for (int v = 0; v < 8; ++v) {
        int mr = row0 + khalf * 8 + v;
        C[(long long)mr * ldc + col0 + mlane]      = alpha * acc0[v];
        C[(long long)mr * ldc + col0 + 16 + mlane] = alpha * acc1[v];
    }
}

// ---------------------------------------------------------------------------
// In-place softmax over rows of length 512. One wave32 per row (16 elems/lane).
// ---------------------------------------------------------------------------
__global__ __launch_bounds__(128)
void softmax512_kernel(float* __restrict__ p, int nrows) {
    int row  = blockIdx.x * 4 + (threadIdx.x >> 5);
    int lane = threadIdx.x & 31;
    if (row >= nrows) return;
    float* r = p + (long long)row * 512;

    float vals[16];
    float mx = -3.4028235e38f;
    #pragma unroll
    for (int i = 0; i < 16; ++i) {
        vals[i] = r[lane + i * 32];
        mx = fmaxf(mx, vals[i]);
    }
    #pragma unroll
    for (int o = 16; o >= 1; o >>= 1) mx = fmaxf(mx, __shfl_xor(mx, o, 32));

    float sum = 0.0f;
    #pragma unroll
    for (int i = 0; i < 16; ++i) {
        vals[i] = expf(vals[i] - mx);
        sum += vals[i];
    }
    #pragma unroll
    for (int o = 16; o >= 1; o >>= 1) sum += __shfl_xor(sum, o, 32);

    float inv = 1.0f / sum;
    #pragma unroll
    for (int i = 0; i < 16; ++i) r[lane + i * 32] = vals[i] * inv;
}

// ---------------------------------------------------------------------------
extern "C" void kernel_launch(void* const* d_in, const int* in_sizes, int n_in,
                              void* d_out, int out_size, void* d_ws, size_t ws_size,
                              hipStream_t stream) {
    const float* emb = (const float*)d_in[0];
    // d_in[1] graph_init: unused — graph is masked to zero (activity = 1/S < 0.1)
    const float* EL  = (const float*)d_in[2];   // [64,256]
    const float* pos = (const float*)d_in[3];   // [1000,256]
    const float* wq  = (const float*)d_in[4];   // [256,256]  (DE, H*DH)
    const float* wk  = (const float*)d_in[5];   // [256,256]
    const float* wv  = (const float*)d_in[6];   // [256,256]
    const float* wo  = (const float*)d_in[7];   // [256,256]  (H*DH, DE)
    // w1/b1/w2/b2/wg1/bg1/wg2/bg2 (d_in[8..15]): dead — outputs they feed are zeroed.

    float* out = (float*)d_out;                 // [entities 131072 | graph 32768 | mask 512]
    float* ws  = (float*)d_ws;

    const long long NX  = (long long)BCONST * SCONST * DCONST;   // 1,048,576
    float* x    = ws;                        // [4096,256]
    float* kbuf = ws + NX;                   // [4096,256]
    float* vbuf = ws + 2 * NX;               // [4096,256]
    float* q    = ws + 3 * NX;               // [64,256]
    float* sc   = q + 64 * 256;              // [64 batches][64,512]
    float* ctx  = sc + 64ll * 64 * 512;      // [512,256]

    const int threads = 256;

    // x = emb + pos
    add_pos_kernel<<<(int)((NX + threads - 1) / threads), threads, 0, stream>>>(
        emb, pos, x, (int)NX);

    // graph + mask outputs are exactly zero
    {
        int ntail = BCONST * MCONST * MCONST + BCONST * MCONST;  // 33280
        zero_f32_kernel<<<(ntail + threads - 1) / threads, threads, 0, stream>>>(
            out + BCONST * MCONST * DCONST, ntail);
    }

    // q = EL @ Wq                              [64,256] x [256,256]
    wmma_gemm_nn<256><<<dim3(2, 4, 1), 128, 0, stream>>>(
        EL, 256, 1, 0, 0,  wq, 1, 0, 0,  q, 256, 1, 0, 0,
        64, 256, 256, 1.0f);

    // k = x @ Wk ; v = x @ Wv                  [4096,256] x [256,256]
    wmma_gemm_nn<256><<<dim3(2, 256, 1), 128, 0, stream>>>(
        x, 256, 1, 0, 0,  wk, 1, 0, 0,  kbuf, 256, 1, 0, 0,
        4096, 256, 256, 1.0f);
    wmma_gemm_nn<256><<<dim3(2, 256, 1), 128, 0, stream>>>(
        x, 256, 1, 0, 0,  wv, 1, 0, 0,  vbuf, 256, 1, 0, 0,
        4096, 256, 256, 1.0f);

    // scores[z=b*8+h] = (q_h @ k_bh^T) / sqrt(32)   [64,32] x [512,32]^T -> [64,512]
    wmma_gemm_nt<256><<<dim3(4, 4, 64), 128, 0, stream>>>(
        q,    256, HCONST, 0,                 DHCONST,            // A: +h*32
        kbuf, HCONST, (long long)SCONST * DCONST, DHCONST,        // Bt: +b*512*256 + h*32
        sc,   512, 1,      (long long)MCONST * SCONST, 0,         // C: +z*64*512
        64, 512, 32, 0.17677669529663687f);

    // softmax over S (rows of 512); 64*64 = 4096 rows
    softmax512_kernel<<<4096 / 4, 128, 0, stream>>>(sc, 4096);

    // ctx[b,m,h,:] = attn_bh @ v_bh            [64,512] x [512,32]
    wmma_gemm_nn<256><<<dim3(1, 4, 64), 128, 0, stream>>>(
        sc,   512, 1,      (long long)MCONST * SCONST, 0,         // A: +z*64*512
        vbuf, HCONST, (long long)SCONST * DCONST, DHCONST,        // B: +b*512*256 + h*32
        ctx,  256, HCONST, (long long)MCONST * DCONST, DHCONST,   // C: +b*64*256 + h*32
        64, 32, 512, 1.0f);

    // entities = ctx @ Wo -> d_out             [512,256] x [256,256]
    wmma_gemm_nn<256><<<dim3(2, 32, 1), 128, 0, stream>>>(
        ctx, 256, 1, 0, 0,  wo, 1, 0, 0,  out, 256, 1, 0, 0,
        512, 256, 256, 1.0f);
}